// AnomalyAttention_43138651521283
// MI455X (gfx1250) — compile-verified
//
#include <hip/hip_runtime.h>
#include <hip/hip_bf16.h>

typedef __attribute__((ext_vector_type(2))) float v2f;
typedef __attribute__((ext_vector_type(8))) float v8f;

constexpr int B_ = 32, L_ = 512, H_ = 8, E_ = 64, S_ = 512;
constexpr int KSTRIDE = 68;    // padded LDS row stride for K rows (floats) -> conflict-free b64 B-frags
constexpr int VSTRIDE = 516;   // padded LDS row stride for transposed V / score rows (floats)
constexpr int SPLIT = 2;       // L row-tile halves per (b,h) -> grid = B*H*2

// LDS layout (floats)
constexpr int KLDS_OFF = 0;                          // 512*68  = 34816
constexpr int VLDS_OFF = KLDS_OFF + S_ * KSTRIDE;    // 64*516  = 33024
constexpr int SCOR_OFF = VLDS_OFF + E_ * VSTRIDE;    // 16*516  = 8256
constexpr int RED_OFF  = SCOR_OFF + 16 * VSTRIDE;    // 128
constexpr int RED2_OFF = RED_OFF + 128;              // 128
constexpr int SMEM_FLOATS = RED2_OFF + 128;          // 76352 floats = 305408 B  (<= 320KB WGP LDS)

__global__ void __launch_bounds__(128)
anomaly_attn_kernel(const float* __restrict__ Q, const float* __restrict__ K,
                    const float* __restrict__ V, float* __restrict__ seriesOut,
                    float* __restrict__ vOut) {
  extern __shared__ float smem[];
  float* klds = smem + KLDS_OFF;
  float* vlds = smem + VLDS_OFF;
  float* scor = smem + SCOR_OFF;
  float* redA = smem + RED_OFF;
  float* redB = smem + RED2_OFF;

  const int tid  = threadIdx.x;
  const int lane = tid & 31;
  const int wave = tid >> 5;
  const int n16  = lane & 15;          // N (or M) index inside a fragment
  const int kb   = (lane >> 4) << 1;   // K base inside a fragment: 0 or 2

  const int half = blockIdx.x & (SPLIT - 1);
  const int bh   = blockIdx.x / SPLIT;
  const int h    = bh & (H_ - 1);
  const int b    = bh / H_;

  const size_t rowStride = (size_t)H_ * E_;  // 512 floats between sequence positions
  const float* kBase = K + ((size_t)b * L_) * rowStride + (size_t)h * E_;
  const float* vBase = V + ((size_t)b * L_) * rowStride + (size_t)h * E_;
  const float* qBase = Q + ((size_t)b * L_) * rowStride + (size_t)h * E_;

  // ---- Stage K (row-major, padded) and V (transposed, padded) into LDS ----
  for (int idx = tid; idx < S_ * (E_ / 4); idx += 128) {
    const int s  = idx >> 4;           // E_/4 == 16 float4 per row
    const int e4 = (idx & 15) << 2;
    const float4 kv = *(const float4*)(kBase + (size_t)s * rowStride + e4);
    *(float4*)(klds + s * KSTRIDE + e4) = kv;   // 272B row stride -> 16B aligned
    const float4 vv = *(const float4*)(vBase + (size_t)s * rowStride + e4);
    vlds[(e4 + 0) * VSTRIDE + s] = vv.x;
    vlds[(e4 + 1) * VSTRIDE + s] = vv.y;
    vlds[(e4 + 2) * VSTRIDE + s] = vv.z;
    vlds[(e4 + 3) * VSTRIDE + s] = vv.w;
  }
  __syncthreads();

  constexpr float scale = 0.125f;  // 1/sqrt(E)

  const int rt0 = half * (L_ / 16 / SPLIT);
  for (int rt = rt0; rt < rt0 + (L_ / 16 / SPLIT); ++rt) {
    const int row0 = rt * 16;

    // Per-wave Q tile in A-fragment layout: 16 k-steps of 16x4, pre-scaled.
    v2f qa[16];
    {
      const float* qrow = qBase + (size_t)(row0 + n16) * rowStride;
#pragma unroll
      for (int ks = 0; ks < 16; ++ks) {
        v2f t = *(const v2f*)(qrow + 4 * ks + kb);
        t *= scale;
        qa[ks] = t;
      }
    }

    // ---- Phase 1: scores = scale * Q K^T; waves split the 32 column tiles.
    // Compile-time trip count (8) -> no divergent loop, EXEC stays all-1s
    // around the WMMAs as the ISA requires.
    for (int jj = 0; jj < (S_ / 16) / 4; ++jj) {
      const int j = wave + 4 * jj;
      v8f acc = {};
#pragma unroll
      for (int ks = 0; ks < 16; ++ks) {
        const v2f bf = *(const v2f*)(klds + (j * 16 + n16) * KSTRIDE + 4 * ks + kb);
        acc = __builtin_amdgcn_wmma_f32_16x16x4_f32(false, qa[ks], false, bf,
                                                    (short)0, acc, false, false);
      }
#pragma unroll
      for (int r = 0; r < 8; ++r)
        scor[(r + 8 * (lane >> 4)) * VSTRIDE + j * 16 + n16] = acc[r];
    }
    __syncthreads();

    // ---- Softmax over S in LDS (each thread owns a 64-wide strip of one row) ----
    {
      const int row = tid & 15;
      const int strip = tid >> 4;  // 8 strips
      float* p = scor + row * VSTRIDE + strip * 64;
      float mx = -3.4e38f;
      for (int i = 0; i < 64; ++i) mx = fmaxf(mx, p[i]);
      redA[strip * 16 + row] = mx;
      __syncthreads();
      float rmax = redA[row];
      for (int q = 1; q < 8; ++q) rmax = fmaxf(rmax, redA[q * 16 + row]);
      float sum = 0.f;
      for (int i = 0; i < 64; ++i) {
        const float e = __expf(p[i] - rmax);
        p[i] = e;
        sum += e;
      }
      redB[strip * 16 + row] = sum;
      __syncthreads();
      float rsum = redB[row];
      for (int q = 1; q < 8; ++q) rsum += redB[q * 16 + row];
      const float inv = 1.f / rsum;
      for (int i = 0; i < 64; ++i) p[i] *= inv;
    }
    __syncthreads();

    // ---- Write series tile, coalesced float4 ----
    {
      float* dst = seriesOut + (((size_t)(b * H_ + h) * L_) + row0) * S_;
      for (int f = tid; f < 16 * (S_ / 4); f += 128) {
        const int row = f >> 7;            // S_/4 == 128
        const int s4  = (f & 127) << 2;
        *(float4*)(dst + (size_t)row * S_ + s4) = *(const float4*)(scor + row * VSTRIDE + s4);
      }
    }

    // ---- Phase 2: out = series x V; each wave owns one 16-wide slice of E ----
    {
      v8f acc = {};
#pragma unroll 8
      for (int ks = 0; ks < S_ / 4; ++ks) {
        const v2f a  = *(const v2f*)(scor + n16 * VSTRIDE + 4 * ks + kb);
        const v2f bf = *(const v2f*)(vlds + (wave * 16 + n16) * VSTRIDE + 4 * ks + kb);
        acc = __builtin_amdgcn_wmma_f32_16x16x4_f32(false, a, false, bf,
                                                    (short)0, acc, false, false);
      }
#pragma unroll
      for (int r = 0; r < 8; ++r) {
        const int m = r + 8 * (lane >> 4);
        vOut[((size_t)(b * L_ + row0 + m) * H_ + h) * E_ + wave * 16 + n16] = acc[r];
      }
    }
    __syncthreads();  // scor reused by next row tile
  }
}

// prior[b,h,l,s] = inv_sqrt_2pi/sig * exp(-|l-s|^2 / (2 sig^2)); sig from sigma[b,l,h]
__global__ void __launch_bounds__(128)
anomaly_prior_kernel(const float* __restrict__ sigma, float* __restrict__ priorOut) {
  const int idx = blockIdx.x;        // (b*H + h)*L + l
  const int l   = idx & (L_ - 1);
  const int bh  = idx / L_;
  const int h   = bh & (H_ - 1);
  const int b   = bh / H_;

  float sg = sigma[((size_t)b * L_ + l) * H_ + h];
  sg = 1.f / (1.f + __expf(-5.f * sg)) + 1e-5f;
  const float sig = exp2f(sg * 1.5849625007211562f) - 1.f;    // 3^sg - 1
  const float amp = 0.3989422804014327f / sig;                // 1/sqrt(2*pi)/sig
  const float c   = 1.f / (2.f * sig * sig);

  const int s0 = threadIdx.x << 2;   // 128 threads * 4 = 512
  const float lf = (float)l;
  float4 v;
  {
    float d0 = lf - (float)(s0 + 0);
    float d1 = lf - (float)(s0 + 1);
    float d2 = lf - (float)(s0 + 2);
    float d3 = lf - (float)(s0 + 3);
    v.x = amp * __expf(-d0 * d0 * c);
    v.y = amp * __expf(-d1 * d1 * c);
    v.z = amp * __expf(-d2 * d2 * c);
    v.w = amp * __expf(-d3 * d3 * c);
  }
  *(float4*)(priorOut + (size_t)idx * S_ + s0) = v;
}

extern "C" void kernel_launch(void* const* d_in, const int* in_sizes, int n_in,
                              void* d_out, int out_size, void* d_ws, size_t ws_size,
                              hipStream_t stream) {
  const float* Q  = (const float*)d_in[0];  // [B,L,H,E]
  const float* K  = (const float*)d_in[1];  // [B,L,H,E]
  const float* V  = (const float*)d_in[2];  // [B,L,H,E]
  const float* Sg = (const float*)d_in[3];  // [B,L,H]

  float* out       = (float*)d_out;
  float* vOut      = out;                                  // [B,L,H,E]  = 8388608
  float* seriesOut = out + (size_t)B_ * L_ * H_ * E_;      // [B,H,L,S]  = 67108864
  float* priorOut  = seriesOut + (size_t)B_ * H_ * L_ * S_;

  const size_t shmemBytes = (size_t)SMEM_FLOATS * sizeof(float);
  anomaly_attn_kernel<<<dim3(B_ * H_ * SPLIT), dim3(128), shmemBytes, stream>>>(
      Q, K, V, seriesOut, vOut);
  anomaly_prior_kernel<<<dim3(B_ * H_ * L_), dim3(128), 0, stream>>>(Sg, priorOut);
}